// Net_GD_22557168239430
// MI455X (gfx1250) — compile-verified
//
#include <hip/hip_runtime.h>
#include <hip/hip_bf16.h>

#define N_NODES 10000
#define DIM 128
#define N_EDGES 320000
#define STEP_SIZE 0.01f
#define NEG_SLOPE 0.2f
#define NUM_LAYERS 8

typedef __attribute__((ext_vector_type(2))) float v2f;
typedef __attribute__((ext_vector_type(8))) float v8f;

// ---------------------------------------------------------------------------
// h = X @ W  (N x 128) @ (128 x 128), fp32 WMMA 16x16x4.
// block = 256 threads = 8 waves; block owns a 16-row stripe, wave w owns
// columns [16w, 16w+16).  The 8 KB X stripe is staged into LDS ONCE per
// block via the CDNA5 async-copy path (global_load_async_to_lds_b128,
// ASYNCcnt), then all 8 waves read their A-fragments from LDS.
// K loop: 32 chained V_WMMA_F32_16X16X4_F32.
// ---------------------------------------------------------------------------
__global__ void gemm_h_kernel(const float* __restrict__ X,
                              const float* __restrict__ W,
                              float* __restrict__ H) {
    __shared__ __align__(16) float xs[16 * DIM];   // 8 KB stripe

    const int tid  = threadIdx.x;
    const int lane = tid & 31;
    const int wave = tid >> 5;              // 0..7 -> column tile
    const int row0 = blockIdx.x * 16;       // 625 blocks * 16 = 10000 rows
    const int col0 = wave * 16;
    const int m  = lane & 15;
    const int kk = (lane >> 4) << 1;        // lanes 0-15 -> K=0,1 ; 16-31 -> K=2,3

    // ---- async stage X[row0:row0+16, :] -> LDS (512 x 16B chunks) ----
    {
        const unsigned lds_base = (unsigned)(uintptr_t)xs;   // low 32 bits = LDS offset
        const unsigned long long gbase =
            (unsigned long long)(const void*)(X + (size_t)row0 * DIM);
        #pragma unroll
        for (int c = 0; c < 2; ++c) {
            const int chunk = tid + c * 256;                 // 0..511
            unsigned ldsa = lds_base + chunk * 16;
            unsigned long long ga = gbase + (unsigned long long)chunk * 16;
            asm volatile("global_load_async_to_lds_b128 %0, %1, off"
                         :: "v"(ldsa), "v"(ga) : "memory");
        }
        asm volatile("s_wait_asynccnt 0x0" ::: "memory");
    }
    __syncthreads();

    v8f c = {};
    const float* xrow = xs + (size_t)m * DIM;   // LDS reads (ds_load_b64)
    for (int k = 0; k < DIM; k += 4) {
        v2f a, b;
        // A fragment (16x4, ISA layout): VGPR0=K{0|2}, VGPR1=K{1|3}
        a[0] = xrow[k + kk];
        a[1] = xrow[k + kk + 1];
        // B fragment (4x16): row K striped across lanes
        b[0] = W[(k + kk) * DIM + col0 + m];
        b[1] = W[(k + kk + 1) * DIM + col0 + m];
        c = __builtin_amdgcn_wmma_f32_16x16x4_f32(
                /*neg_a=*/false, a, /*neg_b=*/false, b,
                /*c_mod=*/(short)0, c, /*reuse_a=*/false, /*reuse_b=*/false);
    }
    // D layout: VGPR g -> M = g (lanes 0-15) or M = 8+g (lanes 16-31), N = lane&15
    const int rbase = row0 + ((lane >> 4) << 3);
    const int col = col0 + m;
    #pragma unroll
    for (int g = 0; g < 8; ++g)
        H[(size_t)(rbase + g) * DIM + col] = c[g];
}

// ---------------------------------------------------------------------------
// s_src[n] = h[n]·att_src ; s_dst[n] = h[n]·att_dst  (one wave per node)
// ---------------------------------------------------------------------------
__global__ void dots_kernel(const float* __restrict__ H,
                            const float* __restrict__ att_src,
                            const float* __restrict__ att_dst,
                            float* __restrict__ s_src,
                            float* __restrict__ s_dst) {
    const int lane = threadIdx.x & 31;
    const int node = (blockIdx.x * blockDim.x + threadIdx.x) >> 5;
    if (node >= N_NODES) return;
    const float* hp = H + (size_t)node * DIM;
    float a = 0.f, b = 0.f;
    #pragma unroll
    for (int j = lane; j < DIM; j += 32) {
        float hv = hp[j];
        a += hv * att_src[j];
        b += hv * att_dst[j];
    }
    #pragma unroll
    for (int o = 16; o > 0; o >>= 1) {
        a += __shfl_down(a, o, 32);
        b += __shfl_down(b, o, 32);
    }
    if (lane == 0) { s_src[node] = a; s_dst[node] = b; }
}

// ordered-uint encoding of float for monotone atomicMax
__device__ __forceinline__ unsigned f2ord(float v) {
    unsigned u = __float_as_uint(v);
    return (u & 0x80000000u) ? ~u : (u | 0x80000000u);
}
__device__ __forceinline__ float ord2f(unsigned u) {
    u = (u & 0x80000000u) ? (u & 0x7FFFFFFFu) : ~u;
    return __uint_as_float(u);
}

__global__ void zero_kernel(float* __restrict__ agg,
                            float* __restrict__ denom,
                            unsigned* __restrict__ mkey) {
    const int i = blockIdx.x * blockDim.x + threadIdx.x;
    if (i < N_NODES * DIM) agg[i] = 0.f;
    if (i < N_NODES) { denom[i] = 0.f; mkey[i] = 0u; }
}

// pass 1: edge logits + segment max over dst
__global__ void edge1_kernel(const int* __restrict__ src,
                             const int* __restrict__ dst,
                             const float* __restrict__ s_src,
                             const float* __restrict__ s_dst,
                             float* __restrict__ evals,
                             unsigned* __restrict__ mkey) {
    const int e = blockIdx.x * blockDim.x + threadIdx.x;
    if (e >= N_EDGES) return;
    const int s = src[e], d = dst[e];
    float v = s_src[s] + s_dst[d];
    v = v > 0.f ? v : NEG_SLOPE * v;
    evals[e] = v;
    atomicMax(mkey + d, f2ord(v));
}

// pass 2: softmax weights + weighted aggregation (one wave per edge)
__global__ void edge2_kernel(const int* __restrict__ src,
                             const int* __restrict__ dst,
                             const float* __restrict__ evals,
                             const unsigned* __restrict__ mkey,
                             const float* __restrict__ H,
                             float* __restrict__ denom,
                             float* __restrict__ agg) {
    const int lane = threadIdx.x & 31;
    const int e = (blockIdx.x * blockDim.x + threadIdx.x) >> 5;
    if (e >= N_EDGES) return;
    const int s = src[e], d = dst[e];
    const float w = __expf(evals[e] - ord2f(mkey[d]));
    if (lane == 0) atomicAdd(denom + d, w);
    const float* hp = H + (size_t)s * DIM;
    float* ap = agg + (size_t)d * DIM;
    #pragma unroll
    for (int j = lane; j < DIM; j += 32)
        atomicAdd(ap + j, w * hp[j]);
}

// ---------------------------------------------------------------------------
// per-node: u = agg/denom ; grad = 2*A@u + b ; xx = u - alpha*grad + th*(z_old-u)
// then z <- u.  Block = 256 threads = 8 waves, one node per block.
// Streams 64 KB of A per node with float4 loads (HBM-bound: the hot loop).
// ---------------------------------------------------------------------------
__global__ void update_kernel(const float* __restrict__ agg,
                              const float* __restrict__ denom,
                              const float* __restrict__ A,
                              const float* __restrict__ b,
                              const float* __restrict__ theta,
                              float alpha,
                              float* __restrict__ xx,
                              float* __restrict__ z) {
    __shared__ float u[DIM];
    const int node = blockIdx.x;
    const int tid = threadIdx.x;
    const int lane = tid & 31, wave = tid >> 5;

    const float dn = denom[node] + 1e-16f;
    if (tid < DIM) u[tid] = agg[(size_t)node * DIM + tid] / dn;
    __syncthreads();

    const float th = theta[0];
    const float* An = A + (size_t)node * DIM * DIM;

    for (int t = 0; t < 16; ++t) {
        const int row = wave * 16 + t;
        const float4* arow = (const float4*)(An + (size_t)row * DIM);
        if (t + 1 < 16)   // stream-prefetch next row (global_prefetch_b8)
            __builtin_prefetch(An + (size_t)(row + 1) * DIM + lane * 4, 0, 0);
        const float4 av = arow[lane];
        float acc = av.x * u[lane * 4 + 0] + av.y * u[lane * 4 + 1] +
                    av.z * u[lane * 4 + 2] + av.w * u[lane * 4 + 3];
        #pragma unroll
        for (int o = 16; o > 0; o >>= 1) acc += __shfl_down(acc, o, 32);
        if (lane == 0) {
            const float ur = u[row];
            const float zold = z[(size_t)node * DIM + row];
            const float grad = 2.f * acc + b[(size_t)node * DIM + row];
            xx[(size_t)node * DIM + row] = ur - alpha * grad + th * (zold - ur);
            z[(size_t)node * DIM + row] = ur;
        }
    }
}

__global__ void copy2_kernel(const float* __restrict__ x,
                             float* __restrict__ xx,
                             float* __restrict__ z) {
    const int i = blockIdx.x * blockDim.x + threadIdx.x;
    if (i < N_NODES * DIM) { const float v = x[i]; xx[i] = v; z[i] = v; }
}

__global__ void out_kernel(const float* __restrict__ xx, float* __restrict__ out) {
    const int i = blockIdx.x * blockDim.x + threadIdx.x;
    if (i < N_NODES * DIM) out[i] = xx[i];
    if (i == 0) ((int*)out)[N_NODES * DIM] = NUM_LAYERS * N_EDGES;  // comm_cost
}

// ---------------------------------------------------------------------------
extern "C" void kernel_launch(void* const* d_in, const int* in_sizes, int n_in,
                              void* d_out, int out_size, void* d_ws, size_t ws_size,
                              hipStream_t stream) {
    const float* x        = (const float*)d_in[0];
    const int*   ei       = (const int*)d_in[1];     // [2, E]
    const float* A        = (const float*)d_in[2];
    const float* b        = (const float*)d_in[3];
    const float* W        = (const float*)d_in[4];
    const float* att_src  = (const float*)d_in[5];
    const float* att_dst  = (const float*)d_in[6];
    const float* theta    = (const float*)d_in[7];
    float* out = (float*)d_out;

    const int* src = ei;
    const int* dst = ei + N_EDGES;

    // workspace layout (floats)
    float* ws = (float*)d_ws;
    float*    xx    = ws;                          // N*D
    float*    z     = xx   + (size_t)N_NODES * DIM;// N*D
    float*    h     = z    + (size_t)N_NODES * DIM;// N*D
    float*    agg   = h    + (size_t)N_NODES * DIM;// N*D
    float*    s_src = agg  + (size_t)N_NODES * DIM;// N
    float*    s_dst = s_src + N_NODES;             // N
    float*    denom = s_dst + N_NODES;             // N
    unsigned* mkey  = (unsigned*)(denom + N_NODES);// N
    float*    evals = (float*)(mkey + N_NODES);    // E

    const int flatND = N_NODES * DIM;                 // 1,280,000
    const int gFlat  = (flatND + 255) / 256;          // 5000
    const int gEdgeT = (N_EDGES + 255) / 256;         // 1250
    const int gEdgeW = (N_EDGES * 32) / 256;          // 40000
    const int gDots  = (N_NODES * 32 + 255) / 256;    // 1250

    copy2_kernel<<<gFlat, 256, 0, stream>>>(x, xx, z);

    for (int i = 0; i < NUM_LAYERS; ++i) {
        const float alpha = STEP_SIZE / sqrtf((float)(i + 1));
        gemm_h_kernel<<<N_NODES / 16, 256, 0, stream>>>(xx, W, h);
        dots_kernel<<<gDots, 256, 0, stream>>>(h, att_src, att_dst, s_src, s_dst);
        zero_kernel<<<gFlat, 256, 0, stream>>>(agg, denom, mkey);
        edge1_kernel<<<gEdgeT, 256, 0, stream>>>(src, dst, s_src, s_dst, evals, mkey);
        edge2_kernel<<<gEdgeW, 256, 0, stream>>>(src, dst, evals, mkey, h, denom, agg);
        update_kernel<<<N_NODES, 256, 0, stream>>>(agg, denom, A, b, theta, alpha, xx, z);
    }

    out_kernel<<<gFlat, 256, 0, stream>>>(xx, out);
    (void)in_sizes; (void)n_in; (void)out_size; (void)ws_size;
}